// LaneATT_Test_66597762892597
// MI455X (gfx1250) — compile-verified
//
#include <hip/hip_runtime.h>
#include <stdint.h>

#define N        8192
#define NOFF     72
#define PCOLS    77
#define NSTRIPS  71
#define TI       32
#define TJ       128
#define SWORDS   256   // N / 32 bitmask words per row
#define LPAD     73    // LDS row stride in floats (73 % 64 = 9, gcd(9,64)=1 -> conflict free)

// ---------------- gfx1250 async global->LDS helpers ----------------
static __device__ __forceinline__ unsigned lds_addr_u32(const void* p) {
    // generic LDS address = {aperture[63:32], offset[31:0]}; low 32 bits are the LDS byte offset
    return (unsigned)(unsigned long long)p;
}
static __device__ __forceinline__ void async_ld_b32(unsigned lds_off, const float* g) {
    asm volatile("global_load_async_to_lds_b32 %0, %1, off"
                 :: "v"(lds_off), "v"(g) : "memory");
}
static __device__ __forceinline__ void wait_async_all() {
    asm volatile("s_wait_asynccnt 0" ::: "memory");
}

// ---------------- 1) bitonic sort by descending score ----------------
__global__ __launch_bounds__(1024)
void nms_sort(const float* __restrict__ scores, unsigned* __restrict__ order) {
    __shared__ unsigned long long keys[N];           // 64 KB LDS (WGP has 320 KB)
    const int tid = threadIdx.x;
    for (int s = tid; s < N; s += 1024) {
        unsigned b = __float_as_uint(scores[s]);     // scores in [0,1): bit order == value order
        keys[s] = ((unsigned long long)b << 32) | (unsigned)(8191 - s); // ties -> smaller idx first
    }
    __syncthreads();
    for (int k = 2; k <= N; k <<= 1) {
        for (int j = k >> 1; j > 0; j >>= 1) {
            for (int m = tid; m < N; m += 1024) {
                int p = m ^ j;
                if (p > m) {
                    unsigned long long a = keys[m], b = keys[p];
                    bool desc = ((m & k) == 0);
                    if ((a < b) == desc) { keys[m] = b; keys[p] = a; }
                }
            }
            __syncthreads();
        }
    }
    for (int s = tid; s < N; s += 1024)
        order[s] = 8191u - (unsigned)(keys[s] & 0xffffffffu);
}

// ---------------- 2) gather sorted rows, compute start/end ----------------
__global__ void nms_gather(const float* __restrict__ prop, const unsigned* __restrict__ order,
                           float* __restrict__ xs, int* __restrict__ si, int* __restrict__ ei) {
    const int s = blockIdx.x;
    __shared__ unsigned so;
    if (threadIdx.x == 0) so = order[s];
    __syncthreads();
    const float* row = prop + (size_t)so * PCOLS;
    const int t = threadIdx.x;
    if (t < NOFF) xs[(size_t)s * NOFF + t] = row[5 + t];
    if (t == 0) {
        int st = __float2int_rn(row[2] * (float)NSTRIPS);   // rn == round-half-even == jnp.round
        int ln = __float2int_rn(row[4]);
        int en = st + ln - 1; if (en > NSTRIPS) en = NSTRIPS;
        si[s] = st; ei[s] = en;
    }
}

// ---------------- 3) pairwise suppression bitmask ----------------
__global__ __launch_bounds__(256)
void nms_supp(const float* __restrict__ xs, const int* __restrict__ si, const int* __restrict__ ei,
              const int* __restrict__ thrp, unsigned* __restrict__ S) {
    __shared__ float lxi[TI * LPAD];
    __shared__ float lxj[TJ * LPAD];
    __shared__ int lsi[TI], lei[TI], lsj[TJ], lej[TJ];
    const int tid = threadIdx.x;
    const int i0 = blockIdx.y * TI, j0 = blockIdx.x * TJ;

    // async-stage both xs tiles into LDS (ASYNCcnt path)
    const float* gi = xs + (size_t)i0 * NOFF;
    for (int t = tid; t < TI * NOFF; t += 256) {
        int r = t / NOFF, c = t - r * NOFF;
        async_ld_b32(lds_addr_u32(&lxi[r * LPAD + c]), gi + r * NOFF + c);
    }
    const float* gj = xs + (size_t)j0 * NOFF;
    for (int t = tid; t < TJ * NOFF; t += 256) {
        int r = t / NOFF, c = t - r * NOFF;
        async_ld_b32(lds_addr_u32(&lxj[r * LPAD + c]), gj + r * NOFF + c);
    }
    if (tid < TI) { lsi[tid] = si[i0 + tid]; lei[tid] = ei[i0 + tid]; }
    if (tid < TJ) { lsj[tid] = si[j0 + tid]; lej[tid] = ei[j0 + tid]; }
    wait_async_all();
    __syncthreads();

    const float thrf = (float)(*thrp);
    const int lane = tid & 31, wid = tid >> 5;
    for (int item = wid; item < TI * 4; item += 8) {      // uniform per wave -> ballot safe
        const int il = item >> 2, wj = item & 3;
        const int jl = wj * 32 + lane;
        const int ps = max(lsi[il], lsj[jl]);
        const int pe = min(lei[il], lej[jl]);
        bool sup = false;
        if (pe >= ps) {
            float d = 0.f;
            const float* a = &lxi[il * LPAD];
            const float* b = &lxj[jl * LPAD];
            for (int k = ps; k <= pe; ++k) d += fabsf(a[k] - b[k]);
            sup = d < thrf * (float)(pe - ps + 1);        // d/cnt < thr
        }
        unsigned mask = (unsigned)__ballot(sup);
        if (lane == 0) S[(size_t)(i0 + il) * SWORDS + (j0 >> 5) + wj] = mask;
    }
}

// ---------------- 4) single-wave greedy scan (wave-synchronous, early exit) ----------------
__global__ void nms_scan(const unsigned* __restrict__ S, const int* __restrict__ topkp,
                         unsigned* __restrict__ keptws, unsigned* __restrict__ cntws) {
    __shared__ unsigned remS[SWORDS];
    volatile unsigned* rem = remS;
    const int lane = threadIdx.x;                 // launched with exactly 32 threads
    for (int r = 0; r < 8; ++r) rem[lane * 8 + r] = 0u;
    int topkL = *topkp; if (topkL < 0) topkL = 0; if (topkL > 63) topkL = 63;
    int count = 0;
    for (int i = 0; i < N; ++i) {
        if (count >= topkL) break;                // later keeps can't affect the output
        unsigned w = rem[i >> 5];                 // same address all lanes -> uniform
        if ((w >> (i & 31)) & 1u) continue;
        if (lane == 0) keptws[count] = (unsigned)i;
        ++count;
        const uint4* row = (const uint4*)(S + (size_t)i * SWORDS);
        uint4 a = row[lane * 2 + 0];
        uint4 b = row[lane * 2 + 1];
        rem[lane * 8 + 0] |= a.x; rem[lane * 8 + 1] |= a.y;
        rem[lane * 8 + 2] |= a.z; rem[lane * 8 + 3] |= a.w;
        rem[lane * 8 + 4] |= b.x; rem[lane * 8 + 5] |= b.y;
        rem[lane * 8 + 6] |= b.z; rem[lane * 8 + 7] |= b.w;
    }
    if (lane == 0) *cntws = (unsigned)count;
}

// ---------------- 5) emit kept rows + num_kept ----------------
__global__ void nms_out(const float* __restrict__ prop, const float* __restrict__ scores,
                        const unsigned* __restrict__ order, const unsigned* __restrict__ keptws,
                        const unsigned* __restrict__ cntws, const int* __restrict__ topkp,
                        float* __restrict__ out) {
    int topkL = *topkp; if (topkL < 0) topkL = 0; if (topkL > 63) topkL = 63;
    const int cnt = (int)*cntws;
    const int total = topkL * (PCOLS + 1);
    for (int e = threadIdx.x; e < total; e += blockDim.x) {
        int r = e / (PCOLS + 1), c = e - r * (PCOLS + 1);
        float v = 0.f;
        if (r < cnt) {
            unsigned orig = order[keptws[r]];
            v = (c < PCOLS) ? prop[(size_t)orig * PCOLS + c] : scores[orig];
        }
        out[e] = v;
    }
    if (threadIdx.x == 0) out[total] = (float)cnt;   // num_kept (already min(kept, topk))
}

extern "C" void kernel_launch(void* const* d_in, const int* in_sizes, int n_in,
                              void* d_out, int out_size, void* d_ws, size_t ws_size,
                              hipStream_t stream) {
    (void)in_sizes; (void)n_in; (void)out_size; (void)ws_size;
    const float* prop   = (const float*)d_in[0];
    const float* scores = (const float*)d_in[1];
    const int*   thrp   = (const int*)d_in[2];
    const int*   topkp  = (const int*)d_in[3];

    char* w = (char*)d_ws;
    unsigned* order = (unsigned*)w;  w += (size_t)N * 4;
    int*      si    = (int*)w;       w += (size_t)N * 4;
    int*      ei    = (int*)w;       w += (size_t)N * 4;
    float*    xs    = (float*)w;     w += (size_t)N * NOFF * 4;
    unsigned* S     = (unsigned*)w;  w += (size_t)N * SWORDS * 4;
    unsigned* kept  = (unsigned*)w;  w += 64 * 4;
    unsigned* cnt   = (unsigned*)w;

    nms_sort  <<<1,                 1024, 0, stream>>>(scores, order);
    nms_gather<<<N,                 128,  0, stream>>>(prop, order, xs, si, ei);
    nms_supp  <<<dim3(N / TJ, N / TI), 256, 0, stream>>>(xs, si, ei, thrp, S);
    nms_scan  <<<1,                 32,   0, stream>>>(S, topkp, kept, cnt);
    nms_out   <<<1,                 256,  0, stream>>>(prop, scores, order, kept, cnt, topkp,
                                                       (float*)d_out);
}